// ScaledDotProductAttention_618475290838
// MI455X (gfx1250) — compile-verified
//
#include <hip/hip_runtime.h>
#include <hip/hip_bf16.h>
#include <stdint.h>

typedef float v2f __attribute__((ext_vector_type(2)));
typedef float v8f __attribute__((ext_vector_type(8)));

#define B_    16
#define LQ_   2048
#define LK_   2048
#define D_    64
#define SCALE 0.125f          // 1 / temperature (temperature = 8.0)
#define ROWP  2052            // padded LDS row stride (floats): 2052%64==4 -> conflict-free
#define NW    4               // waves per block
#define SPAN  (LK_ / NW)      // 512 K-columns per wave
#define SMEM_FLOATS (16 * ROWP + 128 + 128 + 16 + 16)

__global__ __launch_bounds__(128)
void ScaledDotProductAttention_618475290838_kernel(
    const float* __restrict__ q, const float* __restrict__ k,
    const float* __restrict__ v, const uint8_t* __restrict__ mask,
    float* __restrict__ out, float* __restrict__ attn)
{
  extern __shared__ float smem[];
  float* Sbuf   = smem;             // [16][ROWP] scaled logits -> probabilities
  float* smax   = smem + 16 * ROWP; // [16][8] partial row max
  float* ssum   = smax + 128;       // [16][8] partial row sum-exp
  float* rowM   = ssum + 128;       // [16] row max
  float* rowInv = rowM + 16;        // [16] 1 / row sum

  const int t    = threadIdx.x;
  const int w    = t >> 5;
  const int lane = t & 31;
  const int lo   = lane & 15;       // M (A) or N (B/C) index
  const int hi   = lane >> 4;       // K-half selector

  const int bid = blockIdx.x;
  const int b   = bid >> 7;         // LQ_/16 = 128 q-tiles per batch
  const int q0  = (bid & 127) << 4;

  // ---------------- Q tile in WMMA f32 A-layout (16 x float2 chunks) -------
  v2f qa[16];
  {
    const float* qrow = q + ((size_t)(b * LQ_ + q0 + lo)) * D_ + 2 * hi;
#pragma unroll
    for (int c = 0; c < 16; ++c)
      qa[c] = *(const v2f*)(qrow + 4 * c);
  }

  // ---------------- Phase 1: S = (Q K^T) * scale, staged in LDS ------------
  const int n_base = w * SPAN;
  for (int jj = 0; jj < SPAN / 16; ++jj) {
    const int n0 = n_base + jj * 16;
    const float* krow = k + ((size_t)(b * LK_ + n0 + lo)) * D_ + 2 * hi;
    v8f acc = {};
#pragma unroll
    for (int c = 0; c < 16; ++c) {
      v2f kb = *(const v2f*)(krow + 4 * c);
      acc = __builtin_amdgcn_wmma_f32_16x16x4_f32(false, qa[c], false, kb,
                                                  (short)0, acc, false, false);
    }
#pragma unroll
    for (int r = 0; r < 8; ++r) {
      const int m = r + 8 * hi;     // C layout: VGPR r -> row r (+8 for hi half)
      Sbuf[m * ROWP + n0 + lo] = acc[r] * SCALE;
    }
  }
  __syncthreads();

  // ------- Phase 2: vectorized mask (uint4 = 16B) + row max / sum-exp ------
  {
    const int row = t >> 3;         // 8 threads per row
    const int r8  = t & 7;
    const int c0  = r8 * 256;
    float* srow = Sbuf + row * ROWP + c0;
    const uint4* mrow =
        (const uint4*)(mask + ((size_t)(b * LQ_ + q0 + row)) * LK_ + c0);
    float mloc = -INFINITY;
    for (int i = 0; i < 16; ++i) {          // 16 x 16 mask bytes = 256 cols
      union { uint4 u; uint8_t byt[16]; } mu;
      mu.u = mrow[i];
      float sarr[16];
      float4* sv = (float4*)sarr;
      float4* sp = (float4*)(srow + i * 16);
      sv[0] = sp[0]; sv[1] = sp[1]; sv[2] = sp[2]; sv[3] = sp[3];
#pragma unroll
      for (int j = 0; j < 16; ++j) {
        if (mu.byt[j]) sarr[j] = -INFINITY;
        mloc = fmaxf(mloc, sarr[j]);
      }
      sp[0] = sv[0]; sp[1] = sv[1]; sp[2] = sv[2]; sp[3] = sv[3];
    }
    float lloc = 0.0f;
    if (mloc > -INFINITY) {
      for (int i = 0; i < 16; ++i) {
        float4* sp = (float4*)(srow + i * 16);
        float4 s0 = sp[0], s1 = sp[1], s2 = sp[2], s3 = sp[3];
        lloc += __expf(s0.x - mloc) + __expf(s0.y - mloc) +
                __expf(s0.z - mloc) + __expf(s0.w - mloc);
        lloc += __expf(s1.x - mloc) + __expf(s1.y - mloc) +
                __expf(s1.z - mloc) + __expf(s1.w - mloc);
        lloc += __expf(s2.x - mloc) + __expf(s2.y - mloc) +
                __expf(s2.z - mloc) + __expf(s2.w - mloc);
        lloc += __expf(s3.x - mloc) + __expf(s3.y - mloc) +
                __expf(s3.z - mloc) + __expf(s3.w - mloc);
      }
    }
    smax[row * 8 + r8] = mloc;
    ssum[row * 8 + r8] = lloc;
  }
  __syncthreads();
  if (t < 16) {
    float M = -INFINITY;
#pragma unroll
    for (int i = 0; i < 8; ++i) M = fmaxf(M, smax[t * 8 + i]);
    float L = 0.0f;
#pragma unroll
    for (int i = 0; i < 8; ++i) {
      float li = ssum[t * 8 + i];
      if (li > 0.0f) L += li * __expf(smax[t * 8 + i] - M);
    }
    rowM[t]   = M;
    rowInv[t] = (L > 0.0f) ? (1.0f / L) : 0.0f;
  }
  __syncthreads();

  // ---------- Phase 3: normalize, write attn (coalesced float4), keep P ----
  {
    float* attn_tile = attn + ((size_t)(b * LQ_ + q0)) * LK_;
    for (int it = 0; it < 64; ++it) {
      const int idx = (it * 128 + t) * 4;      // 16*2048 floats total
      const int row = idx >> 11;
      const int col = idx & 2047;
      float* sp = Sbuf + row * ROWP + col;
      const float M = rowM[row], inv = rowInv[row];
      float4 p;
      p.x = __expf(sp[0] - M) * inv;
      p.y = __expf(sp[1] - M) * inv;
      p.z = __expf(sp[2] - M) * inv;
      p.w = __expf(sp[3] - M) * inv;
      sp[0] = p.x; sp[1] = p.y; sp[2] = p.z; sp[3] = p.w;
      *(float4*)(attn_tile + (size_t)row * LK_ + col) = p;
    }
  }
  __syncthreads();

  // ---------------- Phase 4: O = P V via WMMA ------------------------------
  v8f oacc[4] = {};
  {
    const float* vbase = v + (size_t)b * LK_ * D_;
    for (int c0 = 0; c0 < SPAN; c0 += 4) {
      const int kk = n_base + c0;
      // A = P chunk 16x4: lane row lo, K = kk + 2*hi + r  (b64 LDS load)
      v2f pa = *(const v2f*)(Sbuf + lo * ROWP + kk + 2 * hi);
      const float* vrow0 = vbase + (size_t)(kk + 2 * hi) * D_;
#pragma unroll
      for (int tn = 0; tn < 4; ++tn) {
        v2f vb;
        vb.x = vrow0[tn * 16 + lo];        // B[k=2hi+0][n]
        vb.y = vrow0[D_ + tn * 16 + lo];   // B[k=2hi+1][n]
        oacc[tn] = __builtin_amdgcn_wmma_f32_16x16x4_f32(false, pa, false, vb,
                                                         (short)0, oacc[tn],
                                                         false, false);
      }
    }
  }
  __syncthreads();

  // Partial-O cross-wave reduction through (reused) LDS, then coalesced store
#pragma unroll
  for (int tn = 0; tn < 4; ++tn) {
#pragma unroll
    for (int r = 0; r < 8; ++r) {
      const int m = r + 8 * hi;
      Sbuf[w * 1024 + m * 64 + tn * 16 + lo] = oacc[tn][r];
    }
  }
  __syncthreads();
  {
    float* out_tile = out + ((size_t)(b * LQ_ + q0)) * D_;
    for (int e = t; e < 16 * 64; e += 128) {
      float s = Sbuf[e] + Sbuf[1024 + e] + Sbuf[2048 + e] + Sbuf[3072 + e];
      out_tile[e] = s;   // row-major 16x64 tile, contiguous == coalesced
    }
  }
}

extern "C" void kernel_launch(void* const* d_in, const int* in_sizes, int n_in,
                              void* d_out, int out_size, void* d_ws, size_t ws_size,
                              hipStream_t stream) {
  (void)in_sizes; (void)n_in; (void)out_size; (void)d_ws; (void)ws_size;
  const float*   q    = (const float*)d_in[0];
  const float*   k    = (const float*)d_in[1];
  const float*   v    = (const float*)d_in[2];
  const uint8_t* mask = (const uint8_t*)d_in[3];  // numpy bool_ = 1 byte
  float* out  = (float*)d_out;
  float* attn = out + (size_t)B_ * LQ_ * D_;      // tuple order: (out, attn)

  const size_t smem_bytes = (size_t)SMEM_FLOATS * sizeof(float);
  ScaledDotProductAttention_618475290838_kernel
      <<<B_ * (LQ_ / 16), 128, smem_bytes, stream>>>(q, k, v, mask, out, attn);
}